// StructuredDiscriminator_12558484373699
// MI455X (gfx1250) — compile-verified
//
#include <hip/hip_runtime.h>
#include <math.h>

typedef __attribute__((ext_vector_type(2))) float v2f;
typedef __attribute__((ext_vector_type(8))) float v8f;

#define BATCH    16
#define LSEQ     2048
#define KNBR     15
#define NKER     16
#define TILE     16
#define NTILES   (LSEQ / TILE)   // 128 column tiles
#define NTHREADS 256
#define NWAVES   8               // wave32 -> 8 waves per block

// One workgroup handles one (batch, 16-row tile). Distance slab for those 16
// rows lives in dynamic LDS (16*2048*4 = 128 KB of the 320 KB/WGP).
__global__ __launch_bounds__(NTHREADS)
void rbf_topk_kernel(const float* __restrict__ T, float* __restrict__ out)
{
    extern __shared__ char smemRaw[];
    float* sdist = (float*)smemRaw;          // [16][2048] squared distances

    __shared__ float rowPos[TILE][3];
    __shared__ float sqRow[TILE];
    __shared__ int   sconn[TILE * KNBR];

    const int tid  = threadIdx.x;
    const int lane = tid & 31;
    const int wave = tid >> 5;
    const int half = lane >> 4;              // 0: lanes 0-15, 1: lanes 16-31
    const int lr   = lane & 15;

    const int b  = blockIdx.x / NTILES;
    const int l0 = (blockIdx.x % NTILES) * TILE;

    // ---- Phase 0: stage row-tile positions (pos = tertiary[:,1,:]) --------
    if (tid < TILE) {
        const float* p = T + (size_t)(b * LSEQ + l0 + tid) * 12 + 3;
        float x = p[0], y = p[1], z = p[2];
        rowPos[tid][0] = x; rowPos[tid][1] = y; rowPos[tid][2] = z;
        sqRow[tid] = x*x + y*y + z*z;
    }
    __syncthreads();

    // A fragment (16x4 f32, K=3 padded with 0):
    // lanes 0-15 hold (K0,K1)=(x,y) of row lr; lanes 16-31 hold (K2,K3)=(z,0)
    v2f afrag;
    if (half == 0) { afrag.x = rowPos[lr][0]; afrag.y = rowPos[lr][1]; }
    else           { afrag.x = rowPos[lr][2]; afrag.y = 0.0f; }

    // ---- Phase 1: Gram tiles via V_WMMA_F32_16X16X4_F32 -------------------
    for (int t = wave; t < NTILES; t += NWAVES) {
        const int c0 = t * TILE;
        const float* p = T + (size_t)(b * LSEQ + c0 + lr) * 12 + 3;
        float cx = p[0], cy = p[1], cz = p[2];
        float sqc = cx*cx + cy*cy + cz*cz;     // |col|^2 for N = lr

        v2f bfrag;                              // B (4x16), mirrored layout
        if (half == 0) { bfrag.x = cx; bfrag.y = cy; }
        else           { bfrag.x = cz; bfrag.y = 0.0f; }

        v8f acc = {};
        acc = __builtin_amdgcn_wmma_f32_16x16x4_f32(
            false, afrag, false, bfrag, (short)0, acc, false, false);

        // C/D layout: VGPR v, lanes 0-15 -> (M=v, N=lane); 16-31 -> (M=v+8)
#pragma unroll
        for (int v = 0; v < 8; ++v) {
            int   M  = v + 8 * half;
            float d2 = sqRow[M] + sqc - 2.0f * acc[v];
            d2 = d2 < 0.0f ? 0.0f : d2;
            sdist[M * LSEQ + c0 + lr] = d2;
        }
    }
    __syncthreads();

    // ---- Phase 2: per-row top-15 smallest dist2 (iterative wave argmin) ---
    for (int r = wave; r < TILE; r += NWAVES) {      // rows wave, wave+8
        float* row = sdist + r * LSEQ;
        for (int k = 0; k < KNBR; ++k) {
            float mval = 3.4e38f; int midx = 0;
            for (int c = lane; c < LSEQ; c += 32) {
                float vv = row[c];
                if (vv < mval) { mval = vv; midx = c; }   // earliest-index min
            }
            for (int off = 16; off; off >>= 1) {          // wave32 reduction
                float ov = __shfl_xor(mval, off, 32);
                int   oi = __shfl_xor(midx, off, 32);
                if (ov < mval || (ov == mval && oi < midx)) { mval = ov; midx = oi; }
            }
            if (lane == 0) sconn[r * KNBR + k] = midx;
            row[midx] = 3.4e38f;   // all lanes, same addr/value: mask winner
        }
    }
    __syncthreads();

    // ---- Phase 3: exact neighbor distance + Gaussian RBF expansion --------
    // mu_q = q * sigma, sigma = 20/15  =>  (d - mu_q)/sigma = d/sigma - q
    const float inv_sigma = 15.0f / 20.0f;
    for (int p = tid; p < TILE * KNBR; p += NTHREADS) {
        int r = p / KNBR, k = p % KNBR;
        size_t iGlob = (size_t)b * LSEQ + l0 + r;
        size_t jGlob = (size_t)b * LSEQ + sconn[r * KNBR + k];
        const float* pj = T + jGlob * 12 + 3;
        float dx = rowPos[r][0] - pj[0] + 1e-6f;
        float dy = rowPos[r][1] - pj[1] + 1e-6f;
        float dz = rowPos[r][2] - pj[2] + 1e-6f;
        float ds = sqrtf(dx*dx + dy*dy + dz*dz) * inv_sigma;

        float* dst = out + (iGlob * KNBR + k) * NKER;
#pragma unroll
        for (int q4 = 0; q4 < 4; ++q4) {
            float vq[4];
#pragma unroll
            for (int j = 0; j < 4; ++j) {
                float tt = ds - (float)(q4 * 4 + j);
                vq[j] = __expf(-tt * tt);
            }
            ((float4*)dst)[q4] = make_float4(vq[0], vq[1], vq[2], vq[3]);
        }
    }
}

extern "C" void kernel_launch(void* const* d_in, const int* in_sizes, int n_in,
                              void* d_out, int out_size, void* d_ws, size_t ws_size,
                              hipStream_t stream) {
    const float* T   = (const float*)d_in[0];   // tertiary (N,4,3) f32
    // d_in[1] (indices) is unused by the reference computation.
    float* out = (float*)d_out;                 // (N, 15, 16) f32

    dim3 grid(BATCH * NTILES);                  // 2048 blocks
    size_t shmem = (size_t)TILE * LSEQ * sizeof(float);   // 128 KB dynamic LDS
    rbf_topk_kernel<<<grid, NTHREADS, shmem, stream>>>(T, out);
}